// GATModel_67448166416492
// MI455X (gfx1250) — compile-verified
//
#include <hip/hip_runtime.h>
#include <math.h>

#define SLOPE 0.2f
#define BN_EPS 1e-5f

typedef float v2f __attribute__((ext_vector_type(2)));
typedef float v8f __attribute__((ext_vector_type(8)));

__device__ __forceinline__ float lrelu(float x) { return x >= 0.f ? x : SLOPE * x; }

// monotone float->uint key so atomicMax(uint) == float max (handles negatives)
__device__ __forceinline__ unsigned fkey(float f) {
    unsigned u = __float_as_uint(f);
    return (u & 0x80000000u) ? ~u : (u | 0x80000000u);
}
__device__ __forceinline__ float funkey(unsigned k) {
    return (k & 0x80000000u) ? __uint_as_float(k & 0x7FFFFFFFu) : __uint_as_float(~k);
}

// ---------------- fill kernels ----------------
__global__ void fill_f32(float* p, float v, long long n) {
    long long i = (long long)blockIdx.x * blockDim.x + threadIdx.x;
    if (i < n) p[i] = v;
}
__global__ void fill_u32(unsigned* p, unsigned v, long long n) {
    long long i = (long long)blockIdx.x * blockDim.x + threadIdx.x;
    if (i < n) p[i] = v;
}

// ---------------- fp32 WMMA GEMM:  Y[nrows,NCOL] = act(X[nrows,K] @ W[K,NCOL] + b) ----------------
// one 16-row tile per block; one wave per 16-col tile; blockDim must be (NCOL/16)*32
template <int K, int NCOL, bool RELU, bool BIAS>
__global__ void gemm_wmma(const float* __restrict__ X, const float* __restrict__ W,
                          const float* __restrict__ bias, float* __restrict__ Y, int nrows) {
    __shared__ float Atile[16 * K];
    const int row0 = blockIdx.x * 16;
    const int wave = threadIdx.x >> 5;
    const int lane = threadIdx.x & 31;
    const int nthreads = (NCOL / 16) * 32;

    for (int i = threadIdx.x; i < 16 * K; i += nthreads) {
        int r = i / K, c = i - r * K;
        int gr = row0 + r;
        Atile[i] = (gr < nrows) ? X[(long long)gr * K + c] : 0.f;
    }
    __syncthreads();

    const int col0 = wave * 16;
    const int m    = lane & 15;
    const int kk   = (lane >> 4) << 1;   // 0 or 2

    v8f acc = {};
#pragma unroll
    for (int k0 = 0; k0 < K; k0 += 4) {
        v2f a, b;
        a.x = Atile[m * K + k0 + kk];
        a.y = Atile[m * K + k0 + kk + 1];
        b.x = W[(k0 + kk) * NCOL + col0 + m];
        b.y = W[(k0 + kk + 1) * NCOL + col0 + m];
        acc = __builtin_amdgcn_wmma_f32_16x16x4_f32(false, a, false, b, (short)0, acc,
                                                    false, false);
    }

    const int ccol  = col0 + m;
    const int rbase = (lane >> 4) * 8;
    float bv = BIAS ? bias[ccol] : 0.f;
#pragma unroll
    for (int r = 0; r < 8; ++r) {
        int gr = row0 + rbase + r;
        if (gr < nrows) {
            float v = acc[r] + bv;
            if (RELU) v = lrelu(v);
            Y[(long long)gr * NCOL + ccol] = v;
        }
    }
}

// ---------------- per-node attention coefficients ----------------
template <int H, int CH>
__global__ void att_coef(const float* __restrict__ xl, const float* __restrict__ as_,
                         const float* __restrict__ ad_, float* __restrict__ asrc,
                         float* __restrict__ adst, int n) {
    long long i = (long long)blockIdx.x * blockDim.x + threadIdx.x;
    if (i >= (long long)n * H) return;
    int node = (int)(i / H), h = (int)(i - (long long)node * H);
    const float* xp = xl + (long long)node * 64 + h * CH;
    float s0 = 0.f, s1 = 0.f;
#pragma unroll
    for (int c = 0; c < CH; ++c) {
        float v = xp[c];
        s0 += v * as_[h * CH + c];
        s1 += v * ad_[h * CH + c];
    }
    asrc[i] = s0;
    adst[i] = s1;
}

// ---------------- edge passes (self-loops appended logically) ----------------
__device__ __forceinline__ void edge_sd(const int* ei, long long Eo, long long e, int& s, int& d) {
    if (e < Eo) { s = ei[e]; d = ei[Eo + e]; }
    else        { s = d = (int)(e - Eo); }
}

template <int H>
__global__ void edge_max(const int* __restrict__ ei, long long Eo, long long Et,
                         const float* __restrict__ asrc, const float* __restrict__ adst,
                         unsigned* __restrict__ maxu) {
    long long idx = (long long)blockIdx.x * blockDim.x + threadIdx.x;
    if (idx >= Et * H) return;
    long long e = idx / H; int h = (int)(idx - e * H);
    int s, d; edge_sd(ei, Eo, e, s, d);
    float al = lrelu(asrc[(long long)s * H + h] + adst[(long long)d * H + h]);
    atomicMax(&maxu[(long long)d * H + h], fkey(al));
}

template <int H>
__global__ void edge_sum(const int* __restrict__ ei, long long Eo, long long Et,
                         const float* __restrict__ asrc, const float* __restrict__ adst,
                         const unsigned* __restrict__ maxu, float* __restrict__ sumz) {
    long long idx = (long long)blockIdx.x * blockDim.x + threadIdx.x;
    if (idx >= Et * H) return;
    long long e = idx / H; int h = (int)(idx - e * H);
    int s, d; edge_sd(ei, Eo, e, s, d);
    float al = lrelu(asrc[(long long)s * H + h] + adst[(long long)d * H + h]);
    float m  = funkey(maxu[(long long)d * H + h]);
    atomicAdd(&sumz[(long long)d * H + h], expf(al - m));
}

template <int H, int CH>
__global__ void edge_msg(const int* __restrict__ ei, long long Eo, long long Et,
                         const float* __restrict__ xl, const float* __restrict__ asrc,
                         const float* __restrict__ adst, const unsigned* __restrict__ maxu,
                         const float* __restrict__ sumz, float* __restrict__ accum) {
    long long idx = (long long)blockIdx.x * blockDim.x + threadIdx.x;
    if (idx >= Et * 64) return;
    long long e = idx >> 6;
    int c = (int)(idx & 63);
    int h = c / CH;
    int s, d; edge_sd(ei, Eo, e, s, d);
    float al = lrelu(asrc[(long long)s * H + h] + adst[(long long)d * H + h]);
    float m  = funkey(maxu[(long long)d * H + h]);
    float z  = sumz[(long long)d * H + h];
    float coef = expf(al - m) / z;
    atomicAdd(&accum[(long long)d * 64 + c], xl[(long long)s * 64 + c] * coef);
}

__global__ void add_bias_lrelu(const float* __restrict__ acc, const float* __restrict__ b,
                               float* __restrict__ out, long long n64) {
    long long i = (long long)blockIdx.x * blockDim.x + threadIdx.x;
    if (i >= n64) return;
    int c = (int)(i & 63);
    out[i] = lrelu(acc[i] + b[c]);
}

// ---------------- pooling + BN1 ----------------
__global__ void pool_acc(const float* __restrict__ h, const int* __restrict__ batch,
                         float* __restrict__ sums, float* __restrict__ cnt, int n) {
    long long i = (long long)blockIdx.x * blockDim.x + threadIdx.x;
    if (i >= (long long)n * 64) return;
    int node = (int)(i >> 6), c = (int)(i & 63);
    int g = batch[node];
    atomicAdd(&sums[(long long)g * 64 + c], h[i]);
    if (c == 0) atomicAdd(&cnt[g], 1.0f);
}

__global__ void pool_bn(const float* __restrict__ sums, const float* __restrict__ cnt,
                        const float* __restrict__ g1, const float* __restrict__ be1,
                        const float* __restrict__ rm1, const float* __restrict__ rv1,
                        float* __restrict__ pooled, int G) {
    int i = blockIdx.x * blockDim.x + threadIdx.x;
    if (i >= G * 64) return;
    int g = i >> 6, c = i & 63;
    float mean = sums[i] / fmaxf(cnt[g], 1.0f);
    pooled[i] = (mean - rm1[c]) * (g1[c] * rsqrtf(rv1[c] + BN_EPS)) + be1[c];
}

// ---------------- head: BN2 + linear(32->1) + sigmoid ----------------
__global__ void head_final(const float* __restrict__ f1, const float* __restrict__ g2,
                           const float* __restrict__ be2, const float* __restrict__ rm2,
                           const float* __restrict__ rv2, const float* __restrict__ Wf2,
                           const float* __restrict__ bf2, float* __restrict__ out, int G) {
    int g = blockIdx.x * blockDim.x + threadIdx.x;
    if (g >= G) return;
    float acc = 0.f;
#pragma unroll
    for (int c = 0; c < 32; ++c) {
        float v = (f1[g * 32 + c] - rm2[c]) * (g2[c] * rsqrtf(rv2[c] + BN_EPS)) + be2[c];
        acc += v * Wf2[c];
    }
    acc += bf2[0];
    out[g] = 1.0f / (1.0f + expf(-acc));
}

// ---------------- launcher ----------------
extern "C" void kernel_launch(void* const* d_in, const int* in_sizes, int n_in,
                              void* d_out, int out_size, void* d_ws, size_t ws_size,
                              hipStream_t stream) {
    const float* x     = (const float*)d_in[0];
    const int*   ei    = (const int*)d_in[1];
    const int*   batch = (const int*)d_in[2];
    const float* W_in  = (const float*)d_in[3];
    const float* b_in  = (const float*)d_in[4];
    const float* W1    = (const float*)d_in[5];
    const float* as1   = (const float*)d_in[6];
    const float* ad1   = (const float*)d_in[7];
    const float* b1    = (const float*)d_in[8];
    const float* W2    = (const float*)d_in[9];
    const float* as2   = (const float*)d_in[10];
    const float* ad2   = (const float*)d_in[11];
    const float* b2    = (const float*)d_in[12];
    const float* Wf1   = (const float*)d_in[13];
    const float* bf1   = (const float*)d_in[14];
    const float* Wf2   = (const float*)d_in[15];
    const float* bf2   = (const float*)d_in[16];
    const float* g1    = (const float*)d_in[17];
    const float* be1   = (const float*)d_in[18];
    const float* rm1   = (const float*)d_in[19];
    const float* rv1   = (const float*)d_in[20];
    const float* g2    = (const float*)d_in[21];
    const float* be2   = (const float*)d_in[22];
    const float* rm2   = (const float*)d_in[23];
    const float* rv2   = (const float*)d_in[24];

    const int       N  = in_sizes[2];
    const long long Eo = in_sizes[1] / 2;
    const long long Et = Eo + N;
    const int       G  = out_size;

    char* ws = (char*)d_ws;
    size_t off = 0;
    auto alloc = [&](size_t bytes) -> void* {
        void* p = ws + off;
        off += (bytes + 255) & ~(size_t)255;
        return p;
    };
    float*    h0     = (float*)alloc((size_t)N * 64 * 4);   // proj out / layer2 out
    float*    xl     = (float*)alloc((size_t)N * 64 * 4);   // per-layer linear
    float*    accum  = (float*)alloc((size_t)N * 64 * 4);   // message accumulator
    float*    h1     = (float*)alloc((size_t)N * 64 * 4);   // layer1 out
    float*    asrc   = (float*)alloc((size_t)N * 4 * 4);
    float*    adst   = (float*)alloc((size_t)N * 4 * 4);
    unsigned* maxu   = (unsigned*)alloc((size_t)N * 4 * 4);
    float*    sumz   = (float*)alloc((size_t)N * 4 * 4);
    float*    sums   = (float*)alloc((size_t)G * 64 * 4);
    float*    cnt    = (float*)alloc((size_t)G * 4);
    float*    pooled = (float*)alloc((size_t)G * 64 * 4);
    float*    f1     = (float*)alloc((size_t)G * 32 * 4);
    (void)ws_size; (void)n_in; (void)b2;

    auto nb = [](long long n, int bs) { return (unsigned)((n + bs - 1) / bs); };
    const unsigned NEG_INF_KEY = 0x007FFFFFu;  // fkey(-inf)

    // input projection: h0 = lrelu(x @ W_in + b_in)
    gemm_wmma<128, 64, true, true><<<(N + 15) / 16, 128, 0, stream>>>(x, W_in, b_in, h0, N);

    // ---- GAT layer 1 (H=4, C=16) ----
    gemm_wmma<64, 64, false, false><<<(N + 15) / 16, 128, 0, stream>>>(h0, W1, nullptr, xl, N);
    att_coef<4, 16><<<nb((long long)N * 4, 256), 256, 0, stream>>>(xl, as1, ad1, asrc, adst, N);
    fill_u32<<<nb((long long)N * 4, 256), 256, 0, stream>>>(maxu, NEG_INF_KEY, (long long)N * 4);
    fill_f32<<<nb((long long)N * 4, 256), 256, 0, stream>>>(sumz, 0.f, (long long)N * 4);
    fill_f32<<<nb((long long)N * 64, 256), 256, 0, stream>>>(accum, 0.f, (long long)N * 64);
    edge_max<4><<<nb(Et * 4, 256), 256, 0, stream>>>(ei, Eo, Et, asrc, adst, maxu);
    edge_sum<4><<<nb(Et * 4, 256), 256, 0, stream>>>(ei, Eo, Et, asrc, adst, maxu, sumz);
    edge_msg<4, 16><<<nb(Et * 64, 256), 256, 0, stream>>>(ei, Eo, Et, xl, asrc, adst, maxu, sumz, accum);
    add_bias_lrelu<<<nb((long long)N * 64, 256), 256, 0, stream>>>(accum, b1, h1, (long long)N * 64);

    // ---- GAT layer 2 (H=1, C=64) ----
    gemm_wmma<64, 64, false, false><<<(N + 15) / 16, 128, 0, stream>>>(h1, W2, nullptr, xl, N);
    att_coef<1, 64><<<nb(N, 256), 256, 0, stream>>>(xl, as2, ad2, asrc, adst, N);
    fill_u32<<<nb(N, 256), 256, 0, stream>>>(maxu, NEG_INF_KEY, (long long)N);
    fill_f32<<<nb(N, 256), 256, 0, stream>>>(sumz, 0.f, (long long)N);
    fill_f32<<<nb((long long)N * 64, 256), 256, 0, stream>>>(accum, 0.f, (long long)N * 64);
    edge_max<1><<<nb(Et, 256), 256, 0, stream>>>(ei, Eo, Et, asrc, adst, maxu);
    edge_sum<1><<<nb(Et, 256), 256, 0, stream>>>(ei, Eo, Et, asrc, adst, maxu, sumz);
    edge_msg<1, 64><<<nb(Et * 64, 256), 256, 0, stream>>>(ei, Eo, Et, xl, asrc, adst, maxu, sumz, accum);
    add_bias_lrelu<<<nb((long long)N * 64, 256), 256, 0, stream>>>(accum, d_in ? (const float*)d_in[12] : nullptr, h0, (long long)N * 64);

    // ---- mean pool + BN1 ----
    fill_f32<<<nb((long long)G * 64, 256), 256, 0, stream>>>(sums, 0.f, (long long)G * 64);
    fill_f32<<<nb(G, 256), 256, 0, stream>>>(cnt, 0.f, (long long)G);
    pool_acc<<<nb((long long)N * 64, 256), 256, 0, stream>>>(h0, batch, sums, cnt, N);
    pool_bn<<<nb((long long)G * 64, 256), 256, 0, stream>>>(sums, cnt, g1, be1, rm1, rv1, pooled, G);

    // ---- head MLP ----
    gemm_wmma<64, 32, true, true><<<(G + 15) / 16, 64, 0, stream>>>(pooled, Wf1, bf1, f1, G);
    head_final<<<nb(G, 256), 256, 0, stream>>>(f1, g2, be2, rm2, rv2, Wf2, bf2, (float*)d_out, G);
}